// QuantumLayer_37486474559519
// MI455X (gfx1250) — compile-verified
//
#include <hip/hip_runtime.h>
#include <math.h>

// ⟨Z_i⟩ for RY-embedding + CNOT-chain circuit reduces analytically to
//   out[b,i] = prod_{j<=i} cos(pi * clamp(in[b,j], 0, 1))
// Memory-bound (0.4 FLOP/B, 786 KB total): optimize the data path, not FLOPs.
// gfx1250 path: async DMA global<->LDS (ASYNCcnt) for fully-coalesced b128
// transactions; compute is per-row registers only.

#define NQ 12
#define THREADS 128          // 4 wave32s per block
#define ROWS_PER_BLOCK 128   // one row per thread
#define TILE_FLOATS (ROWS_PER_BLOCK * NQ)   // 1536 floats = 6 KB
#define VEC4_PER_TILE (TILE_FLOATS / 4)     // 384 float4, 3 per thread

__global__ __launch_bounds__(THREADS)
void QuantumLayer_kernel(const float* __restrict__ in,
                         float* __restrict__ out, int B) {
    __shared__ __align__(16) float tile[TILE_FLOATS];

    const int t = threadIdx.x;
    const long long row0 = (long long)blockIdx.x * ROWS_PER_BLOCK;
    const int rows = (B - row0 < ROWS_PER_BLOCK) ? (int)(B - row0) : ROWS_PER_BLOCK;
    const int nvec = (rows * NQ) / 4;                    // float4s this block moves
    const float* __restrict__ gin  = in  + row0 * NQ;
    float* __restrict__       gout = out + row0 * NQ;

    // ---- Phase 1: async DMA global -> LDS, coalesced b128 per lane -------
    #pragma unroll
    for (int k = 0; k < 3; ++k) {
        const int idx4 = t + k * THREADS;
        if (idx4 < nvec) {
            // Low 32 bits of a generic pointer to LDS == byte offset in LDS.
            const unsigned lds_byte = (unsigned)(uintptr_t)(&tile[idx4 * 4]);
            const float* gp = gin + idx4 * 4;
            asm volatile("global_load_async_to_lds_b128 %0, %1, off"
                         :: "v"(lds_byte), "v"(gp)
                         : "memory");
        }
    }
    asm volatile("s_wait_asynccnt 0" ::: "memory");   // ASYNCcnt -> 0
    __syncthreads();

    // ---- Phase 2: per-thread row compute (registers only) ----------------
    float z[NQ];
    if (t < rows) {
        const float4 r0 = *(const float4*)&tile[t * NQ + 0];
        const float4 r1 = *(const float4*)&tile[t * NQ + 4];
        const float4 r2 = *(const float4*)&tile[t * NQ + 8];
        const float x[NQ] = { r0.x, r0.y, r0.z, r0.w,
                              r1.x, r1.y, r1.z, r1.w,
                              r2.x, r2.y, r2.z, r2.w };
        float p = 1.0f;
        #pragma unroll
        for (int i = 0; i < NQ; ++i) {
            const float xi = fminf(fmaxf(x[i], 0.0f), 1.0f);
            p *= cospif(xi);          // cos(pi * clamp(x))
            z[i] = p;                 // prefix product == <Z_i>
        }
    }
    __syncthreads();                  // all tile reads done before reuse

    if (t < rows) {
        *(float4*)&tile[t * NQ + 0] = make_float4(z[0], z[1], z[2],  z[3]);
        *(float4*)&tile[t * NQ + 4] = make_float4(z[4], z[5], z[6],  z[7]);
        *(float4*)&tile[t * NQ + 8] = make_float4(z[8], z[9], z[10], z[11]);
    }
    __syncthreads();                  // results visible block-wide

    // ---- Phase 3: async DMA LDS -> global, coalesced b128 per lane -------
    #pragma unroll
    for (int k = 0; k < 3; ++k) {
        const int idx4 = t + k * THREADS;
        if (idx4 < nvec) {
            const unsigned lds_byte = (unsigned)(uintptr_t)(&tile[idx4 * 4]);
            float* gp = gout + idx4 * 4;
            asm volatile("global_store_async_from_lds_b128 %0, %1, off"
                         :: "v"(gp), "v"(lds_byte)
                         : "memory");
        }
    }
    asm volatile("s_wait_asynccnt 0" ::: "memory");   // drain before endpgm
}

extern "C" void kernel_launch(void* const* d_in, const int* in_sizes, int n_in,
                              void* d_out, int out_size, void* d_ws, size_t ws_size,
                              hipStream_t stream) {
    const float* in  = (const float*)d_in[0];
    float*       out = (float*)d_out;
    const int B = in_sizes[0] / NQ;                       // 8192
    const int grid = (B + ROWS_PER_BLOCK - 1) / ROWS_PER_BLOCK;   // 64 blocks
    QuantumLayer_kernel<<<grid, THREADS, 0, stream>>>(in, out, B);
}